// PointFusion_22978075033747
// MI455X (gfx1250) — compile-verified
//
#include <hip/hip_runtime.h>
#include <math.h>

#define B_ 2
#define L_ 8
#define H_ 240
#define W_ 320
#define N_ (H_*W_)
#define M_ (L_*N_)
#define EPS_ 1e-8f
#define DIST_TH_ 0.05f
#define DOT_TH_ 0.93969262078590831f  /* cos(20 deg) */

// ---------------------------------------------------------------------------
// Map layout: (B, M, 10) float32, fused in place in d_out.
// Row: [pos.xyz, nrm.xyz, col.rgb, conf]
// The zero pass each call is REQUIRED: at scan step s, slot rows for future
// steps must have conf==0 so project_kernel excludes them (reference scan
// semantics), and the harness replays without re-initializing buffers.
// ---------------------------------------------------------------------------

#define N4_ (B_*M_*10/4)   /* 3,072,000 float4 */

__global__ void setup_kernel(float4* __restrict__ map4, int* __restrict__ corr) {
    int i = blockIdx.x * blockDim.x + threadIdx.x;
    if (i < N4_)      map4[i] = make_float4(0.f, 0.f, 0.f, 0.f);
    if (i < B_ * N_)  corr[i] = M_;
}

// Project every map point into camera s; scatter-min map index into pixel
// bucket. atomicMin is order-independent => deterministic.
__global__ void project_kernel(const float* __restrict__ intr,
                               const float* __restrict__ poses,
                               const float* __restrict__ map,
                               int* __restrict__ corr, int s) {
    int i = blockIdx.x * blockDim.x + threadIdx.x;
    int b = blockIdx.y;
    if (i >= M_) return;
    const float* mr = map + ((size_t)b * M_ + i) * 10;
    float conf = mr[9];
    if (!(conf > 0.f)) return;                       // mvalid
    float px = mr[0], py = mr[1], pz = mr[2];
    const float* P = poses + (size_t)(b * L_ + s) * 16;
    float qx = px - P[3], qy = py - P[7], qz = pz - P[11];
    // pc = R^T (p - t)   (row-vector @ R in the reference)
    float pcx = P[0]*qx + P[4]*qy + P[8]*qz;
    float pcy = P[1]*qx + P[5]*qy + P[9]*qz;
    float pcz = P[2]*qx + P[6]*qy + P[10]*qz;
    if (!(pcz > EPS_)) return;
    float fx = intr[b*16+0], fy = intr[b*16+5];
    float cx = intr[b*16+2], cy = intr[b*16+6];
    int ui = (int)rintf(fx * pcx / pcz + cx);        // round-half-even == jnp.round
    int vi = (int)rintf(fy * pcy / pcz + cy);
    if (ui < 0 || ui >= W_ || vi < 0 || vi >= H_) return;
    atomicMin(&corr[b * N_ + vi * W_ + ui], i);
}

// Per-pixel: recompute frame-s features (depth tile staged into LDS via the
// CDNA5 async global->LDS path) and fuse into the map. s==0 -> init rows [0,N).
__global__ void fuse_kernel(const float* __restrict__ rgb,
                            const float* __restrict__ depth,
                            const float* __restrict__ intr,
                            const float* __restrict__ poses,
                            float* __restrict__ map,
                            int* __restrict__ corr, int s) {
    __shared__ float dtile[17 * 18];                 // 17x17 tile, stride 18
    const int b  = blockIdx.z;
    const int lx = threadIdx.x, ly = threadIdx.y;
    const int x0 = blockIdx.x * 16, y0 = blockIdx.y * 16;
    const int wx = x0 + lx, hy = y0 + ly;

    const float fx = intr[b*16+0], fy = intr[b*16+5];
    const float cx = intr[b*16+2], cy = intr[b*16+6];
    const float* P = poses + (size_t)(b * L_ + s) * 16;
    const float R00=P[0],R01=P[1],R02=P[2],t0=P[3];
    const float R10=P[4],R11=P[5],R12=P[6],t1=P[7];
    const float R20=P[8],R21=P[9],R22=P[10],t2=P[11];

    // ---- async-stage depth tile (+1 right/bottom halo, clamped) ----------
    // Edge clamp in the reference only ever needs w-1/h-1 at the image edge,
    // which (W,H multiples of 16) is always in-tile, so a 1-wide halo on the
    // right/bottom suffices.
    const float* dplane = depth + (size_t)(b * L_ + s) * N_;
    unsigned ldsbase = (unsigned)(unsigned long long)(void*)dtile;
    for (int idx = ly * 16 + lx; idx < 17 * 17; idx += 256) {
        int ty = idx / 17, tx = idx % 17;
        int gy = min(y0 + ty, H_ - 1);
        int gx = min(x0 + tx, W_ - 1);
        const float* src = dplane + gy * W_ + gx;
        unsigned dst = ldsbase + (unsigned)(ty * 18 + tx) * 4u;
        asm volatile("global_load_async_to_lds_b32 %0, %1, off"
                     :: "v"(dst), "v"(src) : "memory");
    }
    asm volatile("s_wait_asynccnt 0" ::: "memory");
    __syncthreads();

    // ---- frame features for this pixel ------------------------------------
    struct V3 { float x, y, z; };
    auto mkpw = [&](int gh, int gw, int ty, int tx) -> V3 {
        float d   = dtile[ty * 18 + tx];
        float gxn = ((float)gw - cx) / fx;
        float gyn = ((float)gh - cy) / fy;
        float px = gxn * d, py = gyn * d, pz = d;
        V3 r;
        r.x = R00*px + R01*py + R02*pz + t0;
        r.y = R10*px + R11*py + R12*pz + t1;
        r.z = R20*px + R21*py + R22*pz + t2;
        return r;
    };
    const float d = dtile[ly * 18 + lx];
    const int w1  = min(wx + 1, W_ - 1);
    const int lx1 = lx + (w1 - wx);
    V3 pa = mkpw(hy, w1,     ly, lx1);
    V3 pb = mkpw(hy, w1 - 1, ly, lx1 - 1);
    const int h1  = min(hy + 1, H_ - 1);
    const int ly1 = ly + (h1 - hy);
    V3 qa = mkpw(h1,     wx, ly1,     lx);
    V3 qb = mkpw(h1 - 1, wx, ly1 - 1, lx);
    // Center point == one of the stencil points (bit-identical), pick it:
    V3 pos = (wx < W_ - 1) ? pb : pa;
    float dxx = pa.x-pb.x, dxy = pa.y-pb.y, dxz = pa.z-pb.z;
    float dyx = qa.x-qb.x, dyy = qa.y-qb.y, dyz = qa.z-qb.z;
    float nx = dxy*dyz - dxz*dyy;
    float ny = dxz*dyx - dxx*dyz;
    float nz = dxx*dyy - dxy*dyx;
    float il = 1.0f / (sqrtf(nx*nx + ny*ny + nz*nz) + EPS_);
    nx *= il; ny *= il; nz *= il;
    float gxn = ((float)wx - cx) / fx;
    float gyn = ((float)hy - cy) / fy;
    float alpha = expf(-(gxn*gxn + gyn*gyn) / 0.72f);   // 2*sigma^2, sigma=0.6
    bool valid = d > 0.0f;
    const float* cp = rgb + ((size_t)(b * L_ + s) * N_ + (size_t)(hy * W_ + wx)) * 3;
    float c0 = cp[0], c1 = cp[1], c2 = cp[2];
    const int n = hy * W_ + wx;
    float* mapb = map + (size_t)b * M_ * 10;

    if (s == 0) {
        float* row = mapb + (size_t)n * 10;
        if (valid) {
            row[0]=pos.x; row[1]=pos.y; row[2]=pos.z;
            row[3]=nx;    row[4]=ny;    row[5]=nz;
            row[6]=c0;    row[7]=c1;    row[8]=c2;
            row[9]=alpha;
        } else {
            #pragma unroll
            for (int k = 0; k < 10; ++k) row[k] = 0.0f;
        }
        return;
    }

    int ci = corr[b * N_ + n];
    corr[b * N_ + n] = M_;          // reset bucket for next step's projection
    bool match = false;
    if (valid && ci < M_) {
        const float* mr = mapb + (size_t)ci * 10;
        float cpx=mr[0], cpy=mr[1], cpz=mr[2];
        float cnx=mr[3], cny=mr[4], cnz=mr[5];
        float ccx=mr[6], ccy=mr[7], ccz=mr[8];
        float cw = mr[9];
        float ex = pos.x-cpx, ey = pos.y-cpy, ez = pos.z-cpz;
        float dist = sqrtf(ex*ex + ey*ey + ez*ez);
        float dotp = nx*cnx + ny*cny + nz*cnz;
        if (dist < DIST_TH_ && dotp > DOT_TH_) {
            match = true;
            float wgt = alpha, den = cw + wgt;
            float ax = cw*cnx + wgt*nx;
            float ay = cw*cny + wgt*ny;
            float az = cw*cnz + wgt*nz;
            float nl = 1.0f / (sqrtf(ax*ax + ay*ay + az*az) + EPS_);
            float* orow = mapb + (size_t)ci * 10;   // conflict-free across threads
            orow[0] = (cw*cpx + wgt*pos.x) / den;
            orow[1] = (cw*cpy + wgt*pos.y) / den;
            orow[2] = (cw*cpz + wgt*pos.z) / den;
            orow[3] = ax * nl; orow[4] = ay * nl; orow[5] = az * nl;
            orow[6] = (cw*ccx + wgt*c0) / den;
            orow[7] = (cw*ccy + wgt*c1) / den;
            orow[8] = (cw*ccz + wgt*c2) / den;
            orow[9] = den;
        }
    }
    float* srow = mapb + (size_t)(s * N_ + n) * 10;  // append slot
    if (valid && !match) {
        srow[0]=pos.x; srow[1]=pos.y; srow[2]=pos.z;
        srow[3]=nx;    srow[4]=ny;    srow[5]=nz;
        srow[6]=c0;    srow[7]=c1;    srow[8]=c2;
        srow[9]=alpha;
    } else {
        #pragma unroll
        for (int k = 0; k < 10; ++k) srow[k] = 0.0f;
    }
}

extern "C" void kernel_launch(void* const* d_in, const int* in_sizes, int n_in,
                              void* d_out, int out_size, void* d_ws, size_t ws_size,
                              hipStream_t stream) {
    const float* rgb   = (const float*)d_in[0];
    const float* depth = (const float*)d_in[1];
    const float* intr  = (const float*)d_in[2];
    const float* poses = (const float*)d_in[3];
    float* map = (float*)d_out;
    int*   corr = (int*)d_ws;                 // B*N ints = 614,400 B of scratch

    setup_kernel<<<(N4_ + 255) / 256, 256, 0, stream>>>((float4*)map, corr);

    dim3 fgrid(W_ / 16, H_ / 16, B_), fblk(16, 16);
    fuse_kernel<<<fgrid, fblk, 0, stream>>>(rgb, depth, intr, poses, map, corr, 0);
    for (int s = 1; s < L_; ++s) {
        project_kernel<<<dim3((M_ + 255) / 256, B_), 256, 0, stream>>>(
            intr, poses, map, corr, s);
        fuse_kernel<<<fgrid, fblk, 0, stream>>>(rgb, depth, intr, poses, map, corr, s);
    }
}